// QuantumInspiredAttention_3848290697886
// MI455X (gfx1250) — compile-verified
//
#include <hip/hip_runtime.h>
#include <cstddef>

// QuantumInspiredAttention for MI455X (gfx1250, wave32, WMMA, TDM).
// Pipeline:
//  1) qkv_gemm      : x(4096x256) @ W_qkv(256x768) -> q,k,v in (B,H,T,64) layout   [f32 WMMA]
//  2) mixed_scores  : la0 = (a1*q.kT + b1*k.qT + c1*q.qT)/tau per 16x16 tile       [3 WMMA chains]
//  3) Sinkhorn      : dual-variable updates r_i, c_j via streaming logsumexp over la0 (128MB, L2-resident)
//  4) attnv         : out = exp(la0 - ro_i - co_j) @ v; v staged in LDS via the
//                     Tensor Data Mover (tensor_load_to_lds + s_wait_tensorcnt),
//                     exp fused into the WMMA A-fragment                            [f32 WMMA + TDM]
//  5) proj_gemm     : o(4096x256) @ W_proj(256x256) + b_proj                        [f32 WMMA]
// Workspace requirement: ~151 MB (144 MiB) of f32 scratch.

#define B_   2
#define T_   2048
#define D_   256
#define H_   4
#define HD_  64
#define BH_  (B_ * H_)
#define SCALE_ 0.125f
#define ITERS_ 5
#define CH_  256   // v rows staged per LDS chunk (256*64*4 = 64 KB)

typedef __attribute__((ext_vector_type(2))) float v2f;
typedef __attribute__((ext_vector_type(8))) float v8f;
typedef __attribute__((ext_vector_type(4))) unsigned int u32x4;
typedef __attribute__((ext_vector_type(4))) int i32x4;
typedef __attribute__((ext_vector_type(8))) int i32x8;

// D = A(16x4 f32) * B(4x16 f32) + C(16x16 f32). Native CDNA5 fp32 matrix op.
__device__ __forceinline__ v8f wmma_f32(v2f a, v2f b, v8f c) {
  return __builtin_amdgcn_wmma_f32_16x16x4_f32(false, a, false, b, (short)0, c,
                                               false, false);
}

// ---------------------------------------------------------------------------
// Kernel 1: qkv = x @ W_qkv, scattered into q/k/v buffers as (B*H, T, 64).
// One wave per 16x16 output tile. M=4096, N=768, K=256.
// ---------------------------------------------------------------------------
__global__ void qkv_gemm_kernel(const float* __restrict__ x,
                                const float* __restrict__ w,
                                float* __restrict__ q, float* __restrict__ k,
                                float* __restrict__ v) {
  const int lane = threadIdx.x & 31;
  const int wave = (int)((blockIdx.x * blockDim.x + threadIdx.x) >> 5);
  const int nt = wave % 48;           // 768/16 column tiles
  const int mt = wave / 48;           // 4096/16 row tiles
  const int m0 = mt * 16, n0 = nt * 16;
  const int lm = lane & 15;
  const int hi = lane >> 4;
  const int ks = hi * 2;

  v8f acc = {};
  for (int k0 = 0; k0 < D_; k0 += 4) {
    v2f a = *(const v2f*)(x + (size_t)(m0 + lm) * D_ + k0 + ks);  // contiguous pair
    v2f b;
    b.x = w[(size_t)(k0 + ks) * (3 * D_) + n0 + lm];
    b.y = w[(size_t)(k0 + ks + 1) * (3 * D_) + n0 + lm];
    acc = wmma_f32(a, b, acc);
  }

  const int s   = n0 >> 8;            // 0=q 1=k 2=v
  const int rem = n0 & 255;
  const int h   = rem >> 6;
  const int d0  = rem & 63;
  float* dst = (s == 0) ? q : (s == 1) ? k : v;
#pragma unroll
  for (int g = 0; g < 8; ++g) {
    const int m  = m0 + g + hi * 8;
    const int bb = m >> 11;           // batch
    const int t  = m & (T_ - 1);
    dst[((size_t)(bb * H_ + h) * T_ + t) * HD_ + d0 + lm] = acc[g];
  }
}

// ---------------------------------------------------------------------------
// Kernel 2: la0[i,j] = (a1*(q_i.k_j) + b1*(k_i.q_j) + c1*(q_i.q_j)) / tau
// Fuses scores, phase-rotated transpose term and entanglement into one tile
// pass: three WMMA accumulator chains over K=64.
// ---------------------------------------------------------------------------
__global__ void mixed_scores_kernel(const float* __restrict__ q,
                                    const float* __restrict__ k,
                                    const float* __restrict__ phase,
                                    const float* __restrict__ ent,
                                    const float* __restrict__ gate,
                                    const float* __restrict__ tau,
                                    float* __restrict__ la0) {
  const int lane = threadIdx.x & 31;
  const int wave = (int)((blockIdx.x * blockDim.x + threadIdx.x) >> 5);
  const int bh  = wave >> 14;         // 128*128 tiles per (b,h)
  const int rem = wave & 16383;
  const int it = rem >> 7, jt = rem & 127;
  const int i0 = it * 16, j0 = jt * 16;
  const int h  = bh & (H_ - 1);

  const float ph = phase[h];
  const float cp = cosf(ph), sp = sinf(ph);
  const float g  = 1.0f / (1.0f + expf(-gate[h]));
  const float a1 = ((1.0f - g) + g * 0.5f * (cp + sp)) * SCALE_;
  const float b1 = (g * 0.5f * (cp - sp)) * SCALE_;
  const float c1 = 0.1f * g * ent[h];
  const float itau = 1.0f / tau[0];

  const float* qh = q + (size_t)bh * T_ * HD_;
  const float* kh = k + (size_t)bh * T_ * HD_;
  const int lm = lane & 15;
  const int hi = lane >> 4;
  const int ks = hi * 2;

  v8f s1 = {}, s2 = {}, s3 = {};
#pragma unroll
  for (int d0 = 0; d0 < HD_; d0 += 4) {
    v2f aq = *(const v2f*)(qh + (size_t)(i0 + lm) * HD_ + d0 + ks);
    v2f ak = *(const v2f*)(kh + (size_t)(i0 + lm) * HD_ + d0 + ks);
    v2f bq = *(const v2f*)(qh + (size_t)(j0 + lm) * HD_ + d0 + ks);
    v2f bk = *(const v2f*)(kh + (size_t)(j0 + lm) * HD_ + d0 + ks);
    s1 = wmma_f32(aq, bk, s1);  // S      tile: q_i . k_j
    s2 = wmma_f32(ak, bq, s2);  // S^T    tile: k_i . q_j
    s3 = wmma_f32(aq, bq, s3);  // entang tile: q_i . q_j
  }

  float* out = la0 + (size_t)bh * T_ * T_;
#pragma unroll
  for (int gg = 0; gg < 8; ++gg) {
    const int i = i0 + gg + hi * 8;
    out[(size_t)i * T_ + j0 + lm] =
        (a1 * s1[gg] + b1 * s2[gg] + c1 * s3[gg]) * itau;
  }
}

// ---------------------------------------------------------------------------
// Kernel 3: row logsumexp. One wave per row; coalesced reads; online LSE then
// wave32 shfl_xor tree reduction.  out[i] = lse_j(la0[i,j]-c[j]) - rsub[i]
// ---------------------------------------------------------------------------
__global__ void row_lse_kernel(const float* __restrict__ la0,
                               const float* __restrict__ cvec,
                               const float* __restrict__ rsub,
                               float* __restrict__ out) {
  const int lane = threadIdx.x & 31;
  const int row  = (int)((blockIdx.x * blockDim.x + threadIdx.x) >> 5);  // BH*T rows
  const int bh = row >> 11;
  const int i  = row & (T_ - 1);
  const float* base = la0 + (size_t)bh * T_ * T_ + (size_t)i * T_;
  const float* cb   = cvec + bh * T_;

  float m = -3.0e38f, s = 0.0f;
  for (int j = lane; j < T_; j += 32) {
    __builtin_prefetch(base + j + 256, 0, 0);      // global_prefetch_b8
    const float xv = base[j] - cb[j];
    if (xv > m) { s = s * expf(m - xv) + 1.0f; m = xv; }
    else        { s += expf(xv - m); }
  }
#pragma unroll
  for (int off = 16; off > 0; off >>= 1) {
    const float om = __shfl_xor(m, off, 32);
    const float os = __shfl_xor(s, off, 32);
    const float nm = fmaxf(m, om);
    s = s * expf(m - nm) + os * expf(om - nm);
    m = nm;
  }
  if (lane == 0) {
    float r = m + logf(s);
    if (rsub) r -= rsub[row];
    out[row] = r;
  }
}

// ---------------------------------------------------------------------------
// Kernel 4: column logsumexp. One thread per column; consecutive threads hit
// consecutive addresses -> fully coalesced row-slab streaming of la0.
//   out[j] = lse_i(la0[i,j]-r[i]) - csub[j]
// ---------------------------------------------------------------------------
__global__ void col_lse_kernel(const float* __restrict__ la0,
                               const float* __restrict__ rvec,
                               const float* __restrict__ csub,
                               float* __restrict__ out) {
  const int gid = (int)(blockIdx.x * blockDim.x + threadIdx.x);  // BH*T cols
  const int bh = gid >> 11;
  const int j  = gid & (T_ - 1);
  const float* base = la0 + (size_t)bh * T_ * T_ + j;
  const float* rb   = rvec + bh * T_;

  float m = -3.0e38f, s = 0.0f;
  for (int i = 0; i < T_; ++i) {
    __builtin_prefetch(base + (size_t)(i + 8) * T_, 0, 0);
    const float xv = base[(size_t)i * T_] - rb[i];
    if (xv > m) { s = s * expf(m - xv) + 1.0f; m = xv; }
    else        { s += expf(xv - m); }
  }
  float r = m + logf(s);
  if (csub) r -= csub[gid];
  out[gid] = r;
}

__global__ void combine_kernel(const float* __restrict__ r,
                               const float* __restrict__ Rf,
                               const float* __restrict__ c,
                               const float* __restrict__ Cf,
                               float* __restrict__ ro, float* __restrict__ co) {
  const int i = (int)(blockIdx.x * blockDim.x + threadIdx.x);
  ro[i] = r[i] + 0.5f * Rf[i];
  co[i] = c[i] + 0.5f * Cf[i];
}

__global__ void zero_kernel(float* __restrict__ p) {
  p[blockIdx.x * blockDim.x + threadIdx.x] = 0.0f;
}

// ---------------------------------------------------------------------------
// Kernel 5: out = exp(la0 - ro_i - co_j) @ v.
// Block = 8 waves sharing one (b,h); v is staged into LDS in 64 KB chunks via
// the Tensor Data Mover (wave 0 issues one linear tensor_load_to_lds DMA,
// waits on TENSORcnt, workgroup barrier), so 8 waves re-use each chunk from
// LDS instead of re-streaming 512 KB of v per wave from L2. exp() is fused
// into the WMMA A-fragment build (v_exp_f32 co-executes with XDL WMMA).
// ---------------------------------------------------------------------------
__global__ void attnv_kernel(const float* __restrict__ la0,
                             const float* __restrict__ ro,
                             const float* __restrict__ co,
                             const float* __restrict__ v,
                             float* __restrict__ o) {
  __shared__ alignas(16) float vtile[CH_ * HD_];

  const int lane = threadIdx.x & 31;
  const int wib  = threadIdx.x >> 5;            // wave in block: 0..7
  const int bh   = blockIdx.x >> 4;             // 16 blocks per (b,h)
  const int i0   = ((blockIdx.x & 15) * 8 + wib) * 16;
  const int lm = lane & 15;
  const int hi = lane >> 4;
  const int ks = hi * 2;

  const float* base = la0 + (size_t)bh * T_ * T_;
  const float* cob  = co + bh * T_;
  const float* vh   = v + (size_t)bh * T_ * HD_;
  const float  roi  = ro[bh * T_ + i0 + lm];

  v8f acc0 = {}, acc1 = {}, acc2 = {}, acc3 = {};

  for (int c0 = 0; c0 < T_; c0 += CH_) {
    if (c0) __syncthreads();   // previous chunk fully consumed before overwrite

#if defined(__gfx1250__) &&                                         \
    __has_builtin(__builtin_amdgcn_tensor_load_to_lds) &&           \
    __has_builtin(__builtin_amdgcn_s_wait_tensorcnt)
    // ---- TDM path: one linear 64 KB global->LDS DMA, issued by wave 0 ----
    if (wib == 0) {
      const unsigned long long ga =
          (unsigned long long)(const void*)(vh + (size_t)c0 * HD_);
      const unsigned int ldsoff =
          (unsigned int)(unsigned long long)(const void*)&vtile[0];
      const unsigned int NE = CH_ * HD_;  // 16384 f32 elements
      // D# group 0: count=1 | lds_addr | global_addr(57b) | type=2
      u32x4 g0 = { 1u, ldsoff,
                   (unsigned int)(ga & 0xFFFFFFFFull),
                   (unsigned int)((ga >> 32) & 0x01FFFFFFull) | (2u << 30) };
      // D# group 1: data_size=4B; tensor_dim0=tile_dim0=NE; tensor_dim1=tile_dim1=1
      i32x8 g1 = { (int)0x00020000,                 // data_size=2 (4 bytes)
                   (int)((NE & 0xFFFFu) << 16),     // tensor_dim0[15:0]
                   (int)((NE >> 16) | (1u << 16)),  // tensor_dim0[31:16] | tensor_dim1=1
                   (int)((NE & 0xFFFFu) << 16),     // tile_dim0
                   1,                               // tile_dim1=1, tile_dim2=0
                   (int)NE,                         // tensor_dim0_stride[31:0]
                   0, 0 };
      i32x4 gz = {};
#if __clang_major__ >= 23
      i32x8 gz8 = {};
      __builtin_amdgcn_tensor_load_to_lds(g0, g1, gz, gz, gz8, 0);
#else
      __builtin_amdgcn_tensor_load_to_lds(g0, g1, gz, gz, 0);
#endif
      __builtin_amdgcn_s_wait_tensorcnt(0);
    }
#else
    // ---- fallback: cooperative 128-bit staging through VGPRs ----
    for (int t = threadIdx.x; t < (CH_ * HD_) / 4; t += 256) {
      *(float4*)&vtile[t * 4] =
          *(const float4*)(vh + (size_t)c0 * HD_ + (size_t)t * 4);
    }
#endif
    __syncthreads();

    for (int j0 = c0; j0 < c0 + CH_; j0 += 4) {
      const int kk = j0 + ks;
      v2f lv = *(const v2f*)(base + (size_t)(i0 + lm) * T_ + kk);
      v2f cc = *(const v2f*)(cob + kk);
      v2f a;
      a.x = expf(lv.x - roi - cc.x);
      a.y = expf(lv.y - roi - cc.y);
      const int r0 = (kk - c0) * HD_;
      v2f b0, b1, b2, b3;
      b0.x = vtile[r0 + 0  + lm];  b0.y = vtile[r0 + HD_ + 0  + lm];
      b1.x = vtile[r0 + 16 + lm];  b1.y = vtile[r0 + HD_ + 16 + lm];
      b2.x = vtile[r0 + 32 + lm];  b2.y = vtile[r0 + HD_ + 32 + lm];
      b3.x = vtile[r0 + 48 + lm];  b3.y = vtile[r0 + HD_ + 48 + lm];
      acc0 = wmma_f32(a, b0, acc0);
      acc1 = wmma_f32(a, b1, acc1);
      acc2 = wmma_f32(a, b2, acc2);
      acc3 = wmma_f32(a, b3, acc3);
    }
  }

  const int bb = bh >> 2, h = bh & 3;
#pragma unroll
  for (int g = 0; g < 8; ++g) {
    const int t = i0 + g + hi * 8;
    float* orow = o + (size_t)(bb * T_ + t) * D_ + h * HD_;
    orow[0  + lm] = acc0[g];
    orow[16 + lm] = acc1[g];
    orow[32 + lm] = acc2[g];
    orow[48 + lm] = acc3[g];
  }
}

// ---------------------------------------------------------------------------
// Kernel 6: final projection  out = o(4096x256) @ W_proj(256x256) + b_proj
// ---------------------------------------------------------------------------
__global__ void proj_gemm_kernel(const float* __restrict__ o,
                                 const float* __restrict__ w,
                                 const float* __restrict__ bias,
                                 float* __restrict__ out) {
  const int lane = threadIdx.x & 31;
  const int wave = (int)((blockIdx.x * blockDim.x + threadIdx.x) >> 5);
  const int nt = wave & 15;           // 256/16
  const int mt = wave >> 4;           // 4096/16
  const int m0 = mt * 16, n0 = nt * 16;
  const int lm = lane & 15;
  const int hi = lane >> 4;
  const int ks = hi * 2;

  v8f acc = {};
  for (int k0 = 0; k0 < D_; k0 += 4) {
    v2f a = *(const v2f*)(o + (size_t)(m0 + lm) * D_ + k0 + ks);
    v2f b;
    b.x = w[(size_t)(k0 + ks) * D_ + n0 + lm];
    b.y = w[(size_t)(k0 + ks + 1) * D_ + n0 + lm];
    acc = wmma_f32(a, b, acc);
  }
  const float bn = bias[n0 + lm];
#pragma unroll
  for (int g = 0; g < 8; ++g) {
    const int m = m0 + g + hi * 8;
    out[(size_t)m * D_ + n0 + lm] = acc[g] + bn;
  }
}

// ---------------------------------------------------------------------------
extern "C" void kernel_launch(void* const* d_in, const int* in_sizes, int n_in,
                              void* d_out, int out_size, void* d_ws,
                              size_t ws_size, hipStream_t stream) {
  (void)in_sizes; (void)n_in; (void)out_size; (void)ws_size;
  const float* x     = (const float*)d_in[0];
  const float* Wqkv  = (const float*)d_in[1];
  const float* Wproj = (const float*)d_in[2];
  const float* bproj = (const float*)d_in[3];
  const float* phase = (const float*)d_in[4];
  const float* ent   = (const float*)d_in[5];
  const float* gate  = (const float*)d_in[6];
  const float* tau   = (const float*)d_in[7];
  float* out = (float*)d_out;

  float* ws = (float*)d_ws;
  const size_t NQ = (size_t)BH_ * T_ * HD_;        // 1,048,576 floats
  const size_t NV = (size_t)BH_ * T_;              // 16,384 floats
  float* q   = ws;
  float* k   = q + NQ;
  float* v   = k + NQ;
  float* o   = v + NQ;                              // (B,T,DIM) attn output
  float* la0 = o + NQ;                              // 128 MB, L2-resident
  float* rv  = la0 + (size_t)BH_ * T_ * T_;
  float* cv  = rv + NV;
  float* Rf  = cv + NV;
  float* Cf  = Rf + NV;
  float* ro  = Cf + NV;
  float* co  = ro + NV;

  // 1) QKV projection: 12288 waves (one per 16x16 tile), 8 waves/block.
  qkv_gemm_kernel<<<1536, 256, 0, stream>>>(x, Wqkv, q, k, v);

  // 2) Fused mixed log-scores: 131072 tile-waves.
  mixed_scores_kernel<<<16384, 256, 0, stream>>>(q, k, phase, ent, gate, tau, la0);

  // 3) Sinkhorn dual-variable iterations over the L2-resident la0.
  zero_kernel<<<(int)(NV / 256), 256, 0, stream>>>(cv);
  for (int itn = 0; itn < ITERS_; ++itn) {
    row_lse_kernel<<<(int)(NV / 8), 256, 0, stream>>>(la0, cv, nullptr, rv);
    col_lse_kernel<<<(int)(NV / 256), 256, 0, stream>>>(la0, rv, nullptr, cv);
  }
  // Symmetric final normalization terms.
  row_lse_kernel<<<(int)(NV / 8), 256, 0, stream>>>(la0, cv, rv, Rf);
  col_lse_kernel<<<(int)(NV / 256), 256, 0, stream>>>(la0, rv, cv, Cf);
  combine_kernel<<<(int)(NV / 256), 256, 0, stream>>>(rv, Rf, cv, Cf, ro, co);

  // 4) attn @ v: TDM-staged v in LDS, exp() fused into WMMA A-fragment.
  attnv_kernel<<<128, 256, 0, stream>>>(la0, ro, co, v, o);

  // 5) Output projection + bias.
  proj_gemm_kernel<<<512, 256, 0, stream>>>(o, Wproj, bproj, out);
}